// OutputLayer_89069031785170
// MI455X (gfx1250) — compile-verified
//
#include <hip/hip_runtime.h>

// output[i, :] = input_features[point_ids[i], :]
// M = 1M source rows, N = 2M output rows, F = 64 fp32 features (256 B/row).
//
// Pure gather, 0 FLOPs. Memory-bound: ~512 MB stores + ~256-512 MB gather
// reads + 8 MB ids => ~34-45 us at 23.3 TB/s. Strategy:
//   - 16 lanes per row, one float4 (16 B) per lane => fully coalesced
//     global_load_b128 / global_store_b128 per half-wave (256 B).
//   - Non-temporal (NT) stores: output is write-once streaming; keep the
//     192 MB L2 free for the 256 MB gather working set (avg reuse ~2x).
//   - Prefetch the id stream ahead of the dependent gather
//     (global_prefetch_b8 on gfx1250).

typedef float v4f __attribute__((ext_vector_type(4)));

__global__ __launch_bounds__(256) void OutputLayer_gather_kernel(
    const v4f* __restrict__ src,   // input_features as float4 rows (16 per row)
    const int* __restrict__ ids,   // point_ids (int32)
    v4f* __restrict__ dst,         // output as float4
    int nrows)                     // N
{
    const int tid   = blockIdx.x * 256 + threadIdx.x;
    const int row   = tid >> 4;    // 16 float4 chunks per 64-float row
    const int chunk = tid & 15;
    if (row >= nrows) return;

    // Run the index stream well ahead of the dependent gather loads.
    // Speculative prefetch: OOB addresses are silently dropped on gfx1250.
    __builtin_prefetch(ids + row + 16384, /*rw=*/0, /*locality=*/1);

    const int id = ids[row];       // 16 lanes share one id -> merged 4B read

    // Gather read: regular temporal hint so L2 retains the 256 MB hot set.
    const v4f v = src[(size_t)id * 16 + (size_t)chunk];

    // Streaming write: non-temporal so the 512 MB output doesn't thrash L2.
    __builtin_nontemporal_store(v, &dst[(size_t)row * 16 + (size_t)chunk]);
}

extern "C" void kernel_launch(void* const* d_in, const int* in_sizes, int n_in,
                              void* d_out, int out_size, void* d_ws, size_t ws_size,
                              hipStream_t stream) {
    const float* feats = (const float*)d_in[0];   // (M, 64) fp32
    const int*   ids   = (const int*)d_in[1];     // (N,) int32
    float*       out   = (float*)d_out;           // (N, 64) fp32

    const int N = in_sizes[1];                    // number of output rows
    const long long chunks = (long long)N * 16;   // float4 chunks total
    const int blocks = (int)((chunks + 255) / 256);

    OutputLayer_gather_kernel<<<blocks, 256, 0, stream>>>(
        (const v4f*)feats, ids, (v4f*)out, N);
}